// MyBuddyLoss_64398739636352
// MI455X (gfx1250) — compile-verified
//
#include <hip/hip_runtime.h>

// ---------------- CDNA5 / gfx1250 loss kernels ----------------
// Streaming-reduction design (memory bound: ~345MB @ 23.3TB/s ~ 15us floor).
// Deterministic: fixed block partials + fixed-order final reduction (no float atomics).
// WMMA (V_WMMA_F32_16X16X4_F32, ones-B trick) performs the final exact f32 sums.

typedef float v2f __attribute__((ext_vector_type(2)));
typedef float v8f __attribute__((ext_vector_type(8)));

#define T_FRAMES 4000
#define MELS     80
#define TT       512
#define NB       32
#define SEG      200                 // frames per block in frame_kernel
#define WCH      25                  // frames per wave (8 waves * 25 = 200)
#define GA_TC    40                  // frames per block in ga_kernel
#define FR_GX    20                  // frame_kernel gridDim.x (20*200 = 4000)
#define GA_GX    100                 // ga_kernel gridDim.x (100*40 = 4000)
#define NBLK_FR  (FR_GX*NB)          // 640 frame-partial blocks
#define NBLK_GA  (GA_GX*NB)          // 3200 ga-partial blocks
#define WS_FL    0                   // ws[0..31]   : raw frame lengths
#define WS_TL    32                  // ws[32..63]  : raw token lengths
#define WS_FP    64                  // ws[64..]    : frame partials, 640 x 12
#define WS_GA    (64 + NBLK_FR*12)   // ws[7744..]  : ga partials, 3200

__device__ __forceinline__ float wred(float v) {
    #pragma unroll
    for (int o = 16; o > 0; o >>= 1) v += __shfl_xor(v, o, 32);
    return v;
}
__device__ __forceinline__ float wmax(float v) {
    #pragma unroll
    for (int o = 16; o > 0; o >>= 1) v = fmaxf(v, __shfl_xor(v, o, 32));
    return v;
}

// ---------------- kernel 0: per-batch mask lengths ----------------
__global__ void len_kernel(const float* __restrict__ mask,
                           const float* __restrict__ tok,
                           float* __restrict__ ws) {
    __shared__ float red[128];
    const int b = blockIdx.x, tid = threadIdx.x;
    float s = 0.0f;
    for (int t = tid; t < T_FRAMES; t += 128) s += mask[b * T_FRAMES + t];
    red[tid] = s; __syncthreads();
    for (int o = 64; o > 0; o >>= 1) { if (tid < o) red[tid] += red[tid + o]; __syncthreads(); }
    if (tid == 0) ws[WS_FL + b] = red[0];
    __syncthreads();
    float s2 = 0.0f;
    for (int t = tid; t < TT; t += 128) s2 += tok[b * TT + t];
    red[tid] = s2; __syncthreads();
    for (int o = 64; o > 0; o >>= 1) { if (tid < o) red[tid] += red[tid + o]; __syncthreads(); }
    if (tid == 0) ws[WS_TL + b] = red[0];
}

// ---------------- kernel 1: fused per-frame statistics ----------------
// One wave handles WCH consecutive frames; 32 lanes cover the 80 mel bins
// (slots lane, lane+32, lane+64). Delta term carries prev-frame diff in regs.
__global__ void frame_kernel(const float* __restrict__ pred,
                             const float* __restrict__ targ,
                             const float* __restrict__ mask,
                             float* __restrict__ fpart) {
    const int lane = threadIdx.x & 31;
    const int wave = threadIdx.x >> 5;
    const int b    = blockIdx.y;
    const int ts   = blockIdx.x * SEG + wave * WCH;
    const bool has3 = (lane < 16);

    // per-lane accumulators (wave-reduced at the end)
    float a_d2 = 0, a_l1 = 0, a_pwad = 0, a_pw = 0, a_t2 = 0, a_hb = 0, a_band = 0, a_delta = 0;
    // wave-uniform accumulators (NOT reduced)
    float u_en = 0, u_pk = 0, u_dm = 0, u_mk = 0;

    float pd0 = 0, pd1 = 0, pd2 = 0, pm = 0;
    bool have_prev = (ts > 0);
    const size_t rowbase = ((size_t)b * T_FRAMES + (size_t)ts) * MELS;

    if (have_prev) {   // boundary frame: re-read previous frame once per wave (4% overhead)
        size_t pb = rowbase - MELS;
        float p0 = pred[pb + lane],      p1 = pred[pb + lane + 32];
        float q0 = targ[pb + lane],      q1 = targ[pb + lane + 32];
        float p2 = has3 ? pred[pb + lane + 64] : 0.0f;
        float q2 = has3 ? targ[pb + lane + 64] : 0.0f;
        pd0 = p0 - q0; pd1 = p1 - q1; pd2 = has3 ? (p2 - q2) : 0.0f;
        pm = mask[(size_t)b * T_FRAMES + ts - 1];
    }

    for (int f = 0; f < WCH; ++f) {
        const int t = ts + f;
        const size_t base = rowbase + (size_t)f * MELS;
        float p0 = pred[base + lane],  p1 = pred[base + lane + 32];
        float q0 = targ[base + lane],  q1 = targ[base + lane + 32];
        float p2 = has3 ? pred[base + lane + 64] : 0.0f;
        float q2 = has3 ? targ[base + lane + 64] : 0.0f;
        const float mk = mask[(size_t)b * T_FRAMES + t];

        float d0 = p0 - q0, d1 = p1 - q1, d2 = has3 ? (p2 - q2) : 0.0f;
        float ad0 = fabsf(d0), ad1 = fabsf(d1), ad2 = fabsf(d2);

        a_d2 += mk * (d0 * d0 + d1 * d1 + d2 * d2);
        a_l1 += mk * (ad0 + ad1 + ad2);

        float w0 = 1.0f + 1.5f / (1.0f + __expf(-q0));
        float w1 = 1.0f + 1.5f / (1.0f + __expf(-q1));
        float w2 = has3 ? (1.0f + 1.5f / (1.0f + __expf(-q2))) : 0.0f;
        a_pw   += mk * (w0 + w1 + w2);
        a_pwad += mk * (ad0 * w0 + ad1 * w1 + ad2 * w2);

        a_t2 += mk * (q0 * q0 + q1 * q1 + q2 * q2);
        a_hb += mk * (((lane >= 28) ? ad1 : 0.0f) + (has3 ? ad2 : 0.0f)); // mels 60..79

        // mel-adjacent band term: |d[j]-d[j-1]| for j=1..79 (cross-lane via shuffles)
        float s0 = __shfl_up(d0, 1, 32), s1 = __shfl_up(d1, 1, 32), s2 = __shfl_up(d2, 1, 32);
        float e0 = __shfl(d0, 31, 32),   e1 = __shfl(d1, 31, 32);
        float bb;
        if (lane > 0) bb = fabsf(d0 - s0) + fabsf(d1 - s1) + (has3 ? fabsf(d2 - s2) : 0.0f);
        else          bb = fabsf(d1 - e0) + fabsf(d2 - e1);   // pairs (31,32) and (63,64)
        a_band += mk * bb;

        // energy: |rowsum(diff)| (mean handled in final); peak: |max p - max t|
        float rs = wred(d0 + d1 + d2);
        u_en += mk * fabsf(rs);
        float mp = wmax(fmaxf(fmaxf(p0, p1), has3 ? p2 : -3.0e38f));
        float mq = wmax(fmaxf(fmaxf(q0, q1), has3 ? q2 : -3.0e38f));
        u_pk += mk * fabsf(mp - mq);

        // delta (temporal) term
        if (have_prev) {
            float dm = mk * pm;
            a_delta += dm * (fabsf(d0 - pd0) + fabsf(d1 - pd1) + fabsf(d2 - pd2));
            u_dm += dm;
        }
        have_prev = true;
        pd0 = d0; pd1 = d1; pd2 = d2; pm = mk;
        u_mk += mk;
    }

    a_d2 = wred(a_d2); a_l1 = wred(a_l1); a_pwad = wred(a_pwad); a_pw = wred(a_pw);
    a_t2 = wred(a_t2); a_hb = wred(a_hb); a_band = wred(a_band); a_delta = wred(a_delta);

    __shared__ float lds[8][12];
    if (lane == 0) {
        lds[wave][0] = a_d2;   lds[wave][1] = a_l1;   lds[wave][2] = a_pwad;
        lds[wave][3] = a_pw;   lds[wave][4] = u_en;   lds[wave][5] = u_pk;
        lds[wave][6] = a_t2;   lds[wave][7] = a_delta; lds[wave][8] = u_dm;
        lds[wave][9] = a_band; lds[wave][10] = a_hb;  lds[wave][11] = u_mk;
    }
    __syncthreads();
    if (threadIdx.x < 12) {
        float s = 0.0f;
        #pragma unroll
        for (int w = 0; w < 8; ++w) s += lds[w][threadIdx.x];
        fpart[((size_t)blockIdx.y * gridDim.x + blockIdx.x) * 12 + threadIdx.x] = s;
    }
}

// ---------------- kernel 2: guided attention (dominant 262MB stream) ----------------
__global__ void ga_kernel(const float* __restrict__ attn,
                          const float* __restrict__ ws,
                          float* __restrict__ gpart) {
    const int b = blockIdx.y, tid = threadIdx.x;
    const float fl = fmaxf(ws[WS_FL + b], 1.0f);
    const float tl = fmaxf(ws[WS_TL + b], 1.0f);
    const float invfl = 1.0f / fl, invtl = 1.0f / tl;
    const int tt0 = tid, tt1 = tid + 256;
    const float tp0 = (float)tt0 * invtl, tp1 = (float)tt1 * invtl;
    const float v0 = ((float)tt0 < tl) ? 1.0f : 0.0f;
    const float v1 = ((float)tt1 < tl) ? 1.0f : 0.0f;

    float acc = 0.0f;
    const int tbeg = blockIdx.x * GA_TC;
    for (int f = 0; f < GA_TC; ++f) {
        const int t = tbeg + f;
        if ((float)t >= fl) break;                 // uniform per block; tail rows are invalid
        const size_t base = ((size_t)b * T_FRAMES + (size_t)t) * (size_t)TT;
        if (f + 2 < GA_TC)                          // prefetch 2 rows ahead (in-bounds)
            __builtin_prefetch(&attn[base + 2 * TT], 0, 1);
        float a0 = __builtin_nontemporal_load(&attn[base + tt0]); // read-once, > L2 capacity
        float a1 = __builtin_nontemporal_load(&attn[base + tt1]);
        const float fp = (float)t * invfl;
        float e0 = fp - tp0, e1 = fp - tp1;
        float g0 = 1.0f - __expf(-e0 * e0 * 12.5f);  // 1/(2*sigma^2) = 12.5
        float g1 = 1.0f - __expf(-e1 * e1 * 12.5f);
        acc += v0 * a0 * g0 + v1 * a1 * g1;
    }
    __shared__ float red[256];
    red[tid] = acc; __syncthreads();
    for (int o = 128; o > 0; o >>= 1) { if (tid < o) red[tid] += red[tid + o]; __syncthreads(); }
    if (tid == 0) gpart[(size_t)b * gridDim.x + blockIdx.x] = red[0];
}

// ---------------- WMMA ones-matrix exact-f32 sum: 64 values per issue ----------------
// D(16x16) = A(16x4) x B(4x16, all ones) + C  ==>  D[i][j] = sum_k A[i][k] + C[i][j]
__device__ __forceinline__ float wmma_sum64(const float* __restrict__ p,
                                            int iters, int stride, int off) {
    const int lane = threadIdx.x & 31;
    v8f c = {};
    v2f ones; ones.x = 1.0f; ones.y = 1.0f;
    for (int it = 0; it < iters; ++it) {
        v2f a;
        a.x = p[(it * 64 + lane)      * stride + off];
        a.y = p[(it * 64 + 32 + lane) * stride + off];
        c = __builtin_amdgcn_wmma_f32_16x16x4_f32(
                /*neg_a=*/false, a, /*neg_b=*/false, ones,
                /*c_mod=*/(short)0, c, /*reuse_a=*/false, /*reuse_b=*/false);
    }
    float s = c[0] + c[1] + c[2] + c[3] + c[4] + c[5] + c[6] + c[7]; // rows 0-7 / 8-15 half-sums
    s += __shfl_xor(s, 16, 32);                                      // -> full total, all lanes
    return s;
}

// ---------------- kernel 3: final deterministic combine (single wave) ----------------
__global__ void final_kernel(const float* __restrict__ ws, float* __restrict__ out) {
    const int lane = threadIdx.x;                  // blockDim = 32 -> EXEC all ones for WMMA
    const float* fp = ws + WS_FP;
    const float* gp = ws + WS_GA;

    float s[12];
    #pragma unroll
    for (int k = 0; k < 12; ++k) s[k] = wmma_sum64(fp, NBLK_FR / 64, 12, k);
    const float ga_total = wmma_sum64(gp, NBLK_GA / 64, 1, 0);

    float spec_sum = 0.0f, tw = 0.0f;
    for (int b = 0; b < NB; ++b) {
        float vd = (lane < FR_GX) ? fp[(b * FR_GX + lane) * 12 + 0] : 0.0f;
        float vt = (lane < FR_GX) ? fp[(b * FR_GX + lane) * 12 + 6] : 0.0f;
        float d2b = wred(vd), t2b = wred(vt);
        spec_sum += sqrtf(d2b) / (sqrtf(t2b) + 1.0f);
        tw += fmaxf(ws[WS_FL + b], 1.0f) * fmaxf(ws[WS_TL + b], 1.0f);
    }

    const float mask_sum   = s[11];
    const float num_valid  = mask_sum * (float)MELS + 1e-8f;
    const float num_frames = mask_sum + 1e-8f;
    const float mse    = s[0] / num_valid;
    const float l1     = s[1] / num_valid;
    const float wl1    = s[2] / (s[3] + 1e-8f);
    const float energy = (s[4] * (1.0f / (float)MELS)) / num_frames;
    const float peak   = s[5] / num_frames;
    const float spec   = spec_sum / (float)NB;
    const float ga     = ga_total / (tw + 1e-8f);
    const float delta  = s[7] / (s[8] + 1e-8f);
    const float band   = s[9] / num_frames;
    const float hb     = s[10] / num_frames;
    const float total  = mse + 3.0f * l1 + 2.0f * wl1 + 2.0f * energy + 2.0f * peak
                       + 0.5f * spec + ga + delta + band + hb;
    if (lane == 0) {
        out[0] = total; out[1] = mse;  out[2] = l1;   out[3] = wl1;  out[4] = energy;
        out[5] = peak;  out[6] = spec; out[7] = ga;   out[8] = delta; out[9] = band;
        out[10] = hb;
    }
}

extern "C" void kernel_launch(void* const* d_in, const int* in_sizes, int n_in,
                              void* d_out, int out_size, void* d_ws, size_t ws_size,
                              hipStream_t stream) {
    const float* pred = (const float*)d_in[0];
    const float* targ = (const float*)d_in[1];
    const float* mask = (const float*)d_in[2];
    const float* attn = (const float*)d_in[3];
    const float* tok  = (const float*)d_in[4];
    float* out = (float*)d_out;
    float* ws  = (float*)d_ws;   // needs (64 + 640*12 + 3200)*4 = ~43KB

    len_kernel  <<<NB, 128, 0, stream>>>(mask, tok, ws);
    frame_kernel<<<dim3(FR_GX, NB), 256, 0, stream>>>(pred, targ, mask, ws + WS_FP);
    ga_kernel   <<<dim3(GA_GX, NB), 256, 0, stream>>>(attn, ws, ws + WS_GA);
    final_kernel<<<1, 32, 0, stream>>>(ws, out);
}